// SamplePatches_18013092839745
// MI455X (gfx1250) — compile-verified
//
#include <hip/hip_runtime.h>
#include <stdint.h>

// ---------------- problem constants (from reference) ----------------
#define BATCH     16
#define HL        128
#define WL        128
#define HH        1024
#define WH        1024
#define CH        3
#define LOW_N     (HL * WL)          // 16384
#define NPATCH    16
#define PH        128
#define PW        128
#define SCALE     8                  // HH/HL = WH/WL
#define MAX_TL    (HH - PH)          // 896

#define TOPK_THREADS 256

// gather chunking: one TDM load/store pair per 16 rows of a patch
#define CHUNK_ROWS   16
#define N_CHUNKS     (PH / CHUNK_ROWS)            // 8
#define ROW_ELEMS    (PW * CH)                    // 384 floats per patch row
#define CHUNK_ELEMS  (CHUNK_ROWS * ROW_ELEMS)     // 6144 floats
#define CHUNK_BYTES  (CHUNK_ELEMS * 4)            // 24576 bytes of LDS

// ======================================================================
// Kernel 1: per-batch Gumbel top-k (k = 16) with jax.lax.top_k semantics
//   key[i] = log(att[i]) + gumbel[i]; select 16 maxima in descending
//   order, tie-break lowest index. Emits sampled_attention and the
//   (top,left) patch corners into workspace.
// ======================================================================
__global__ __launch_bounds__(TOPK_THREADS)
void SamplePatches_topk_kernel(const float* __restrict__ att,
                               const float* __restrict__ gum,
                               float* __restrict__ samp_out,   // [B, NPATCH]
                               int*   __restrict__ tl_out)     // [B*NPATCH, 2]
{
    __shared__ int   sel[NPATCH];
    __shared__ float wval[TOPK_THREADS / 32];
    __shared__ int   widx[TOPK_THREADS / 32];

    const int b = blockIdx.x;
    const int t = threadIdx.x;
    const int lane = t & 31;
    const int wave = t >> 5;
    const float* attb = att + (size_t)b * LOW_N;
    const float* gumb = gum + (size_t)b * LOW_N;

    for (int k = 0; k < NPATCH; ++k) {
        // local scan over a strided range, skipping already-selected idxs
        float bv = -__builtin_inff();
        int   bi = 0x7fffffff;
        for (int i = t; i < LOW_N; i += TOPK_THREADS) {
            bool taken = false;
            for (int s = 0; s < k; ++s) taken |= (sel[s] == i);
            if (taken) continue;
            float v = __builtin_logf(attb[i]) + gumb[i];
            if (v > bv || (v == bv && i < bi)) { bv = v; bi = i; }
        }
        // wave32 reduction (tie-break: lower index wins on equal value)
        for (int m = 16; m > 0; m >>= 1) {
            float ov = __shfl_xor(bv, m, 32);
            int   oi = __shfl_xor(bi, m, 32);
            if (ov > bv || (ov == bv && oi < bi)) { bv = ov; bi = oi; }
        }
        if (lane == 0) { wval[wave] = bv; widx[wave] = bi; }
        __syncthreads();
        if (t == 0) {
            float fv = wval[0]; int fi = widx[0];
            for (int w = 1; w < TOPK_THREADS / 32; ++w) {
                float ov = wval[w]; int oi = widx[w];
                if (ov > fv || (ov == fv && oi < fi)) { fv = ov; fi = oi; }
            }
            sel[k] = fi;
            samp_out[b * NPATCH + k] = attb[fi];
            int si = fi >> 7;          // fi / WL
            int sj = fi & (WL - 1);    // fi % WL
            int top  = si * SCALE; if (top  > MAX_TL) top  = MAX_TL;
            int left = sj * SCALE; if (left > MAX_TL) left = MAX_TL;
            tl_out[(b * NPATCH + k) * 2 + 0] = top;
            tl_out[(b * NPATCH + k) * 2 + 1] = left;
        }
        __syncthreads();
    }
}

// ======================================================================
// Kernel 2: patch gather via the Tensor Data Mover.
//   Per (batch, patch, chunk): TDM-load a strided 16x384-elem f32 tile
//   from x_high into LDS, wait TENSORcnt, TDM-store it as a contiguous
//   1-D tile into the output. EXEC is ignored by TDM; one wave per WG.
// ======================================================================
typedef unsigned int u32x4 __attribute__((ext_vector_type(4)));
typedef int          i32x4 __attribute__((ext_vector_type(4)));
typedef int          i32x8 __attribute__((ext_vector_type(8)));

// D# group 0: count=1 (valid, user mode), lds_addr, 57-bit global byte
// address in bits [120:64], type=2 ("image") in bits [127:126].
__device__ __forceinline__ u32x4 tdm_group0(uint32_t lds_addr, uint64_t gaddr) {
    u32x4 g;
    g.x = 1u;
    g.y = lds_addr;
    g.z = (uint32_t)(gaddr & 0xffffffffull);
    g.w = (uint32_t)(gaddr >> 32) | 0x80000000u;   // type = 2
    return g;
}

// D# group 1: workgroup_mask=0 (not in cluster), data_size=2 (4 bytes),
// no atomic barrier / iterate / pad. Dims in data_size units.
__device__ __forceinline__ i32x8 tdm_group1(uint32_t tdim0, uint32_t tdim1,
                                            uint32_t tile0, uint32_t tile1,
                                            uint64_t stride0) {
    i32x8 g;
    g[0] = (int)(2u << 16);                                     // data_size = 4B
    g[1] = (int)((tdim0 & 0xffffu) << 16);                      // tdim0[15:0]
    g[2] = (int)((tdim0 >> 16) | ((tdim1 & 0xffffu) << 16));    // tdim0[31:16] | tdim1[15:0]
    g[3] = (int)((tdim1 >> 16) | (tile0 << 16));                // tdim1[31:16] | tile_dim0
    g[4] = (int)(tile1);                                        // tile_dim1 | tile_dim2(=0)
    g[5] = (int)(uint32_t)(stride0 & 0xffffffffull);            // dim0_stride[31:0]
    g[6] = (int)(uint32_t)((stride0 >> 32) & 0xffffull);        // dim0_stride[47:32] | dim1_stride[15:0]=0
    g[7] = 0;                                                   // dim1_stride[47:16]=0
    return g;
}

__global__ __launch_bounds__(32)
void SamplePatches_gather_kernel(const float* __restrict__ x_high,
                                 float* __restrict__ out,
                                 const int* __restrict__ tl)
{
    __shared__ float buf[CHUNK_ELEMS];                // 24 KB staging tile

    const int gid = blockIdx.x;                       // [0, B*NPATCH*N_CHUNKS)
    const int c   = gid & (N_CHUNKS - 1);
    const int bp  = gid >> 3;                         // (batch, patch) flat
    const int b   = bp >> 4;
    const int top  = tl[bp * 2 + 0];
    const int left = tl[bp * 2 + 1];

    // LDS byte offset: generic shared pointer carries offset in low 32 bits
    const uint32_t lds_off = (uint32_t)(uintptr_t)&buf[0];

    // benign group2/group3: tensor_dim2 = tensor_dim3 = tensor_dim4 = 1,
    // tile_dim3 = tile_dim4 = 0 (unused), strides 0.
    i32x4 g2; g2[0] = 1; g2[1] = 1; g2[2] = 0; g2[3] = 0;
    i32x4 g3; g3[0] = 0; g3[1] = (int)(1u << 16); g3[2] = 0; g3[3] = 0;
    // trailing reserved/extended descriptor group (clang-23 6-arg form): zeros
    i32x8 g4; g4[0] = 0; g4[1] = 0; g4[2] = 0; g4[3] = 0;
              g4[4] = 0; g4[5] = 0; g4[6] = 0; g4[7] = 0;

    const uint64_t src = (uint64_t)(uintptr_t)x_high +
        4ull * ((uint64_t)b * HH * WH * CH +
                (uint64_t)(top + c * CHUNK_ROWS) * WH * CH +
                (uint64_t)left * CH);
    const uint64_t dst = (uint64_t)(uintptr_t)out +
        4ull * ((uint64_t)bp * PH * PW * CH + (uint64_t)c * CHUNK_ELEMS);

    // Load: 2-D tile, CHUNK_ROWS rows of ROW_ELEMS f32, row stride WH*CH.
    {
        u32x4 g0 = tdm_group0(lds_off, src);
        i32x8 g1 = tdm_group1(ROW_ELEMS, CHUNK_ROWS,
                              ROW_ELEMS, CHUNK_ROWS,
                              (uint64_t)(WH * CH));
        __builtin_amdgcn_tensor_load_to_lds(g0, g1, g2, g3, g4, 0);
    }
    __builtin_amdgcn_s_wait_tensorcnt(0);

    // Store: contiguous 1-D tile of CHUNK_ELEMS f32 (tile_dim1 = 0 unused).
    {
        u32x4 g0 = tdm_group0(lds_off, dst);
        i32x8 g1 = tdm_group1(CHUNK_ELEMS, 1,
                              CHUNK_ELEMS, 0,
                              (uint64_t)CHUNK_ELEMS);
        __builtin_amdgcn_tensor_store_from_lds(g0, g1, g2, g3, g4, 0);
    }
    __builtin_amdgcn_s_wait_tensorcnt(0);
}

// ======================================================================
// launch
//   inputs: d_in[0]=x_low (unused), d_in[1]=x_high, d_in[2]=attention,
//           d_in[3]=gumbel_noise
//   output: [B,NPATCH,PH,PW,CH] patches (12,582,912 f32) then
//           [B,NPATCH] sampled_attention (256 f32)
// ======================================================================
extern "C" void kernel_launch(void* const* d_in, const int* in_sizes, int n_in,
                              void* d_out, int out_size, void* d_ws, size_t ws_size,
                              hipStream_t stream) {
    const float* x_high = (const float*)d_in[1];
    const float* att    = (const float*)d_in[2];
    const float* gum    = (const float*)d_in[3];

    float* patches = (float*)d_out;
    float* samp    = patches + (size_t)BATCH * NPATCH * PH * PW * CH;
    int*   tl      = (int*)d_ws;   // [B*NPATCH, 2] patch corners

    SamplePatches_topk_kernel<<<BATCH, TOPK_THREADS, 0, stream>>>(att, gum, samp, tl);
    SamplePatches_gather_kernel<<<BATCH * NPATCH * N_CHUNKS, 32, 0, stream>>>(
        x_high, patches, tl);
}